// GAT_22531398435363
// MI455X (gfx1250) — compile-verified
//
#include <hip/hip_runtime.h>
#include <hip/hip_bf16.h>

// ---------------------------------------------------------------------------
// GATv2 (2 layers) for MI455X / gfx1250, wave32.
// Dense projections use V_WMMA_F32_16X16X4_F32 (full f32 precision), with the
// weight matrix staged transposed + zero-padded in LDS so the fully-unrolled
// K-loop is divergence-free (global_load_b64 A + ds_load_b64 B per WMMA).
// Edge softmax/aggregation are atomic-based scatter passes (memory bound).
// ---------------------------------------------------------------------------

#define DIM_IN   128
#define HID      64
#define HEADS    8
#define DIM_H    8
#define DIM_OUT  40
#define NEG_SLOPE 0.2f

typedef __attribute__((ext_vector_type(2))) float v2f;
typedef __attribute__((ext_vector_type(8))) float v8f;

__device__ __forceinline__ void atomicMaxFloat(float* addr, float val) {
    // monotonic int/uint mapping trick; addr must be initialized to -inf bits.
    if (!(val == val)) return;
    if (val >= 0.0f) {
        atomicMax((int*)addr, __float_as_int(val));
    } else {
        atomicMin((unsigned int*)addr, __float_as_uint(val));
    }
}

// ---------------------------------------------------------------------------
// fill: write a 32-bit pattern
// ---------------------------------------------------------------------------
__global__ void gat_fill_u32(unsigned int* __restrict__ p, unsigned int val,
                             long long n) {
    long long i = (long long)blockIdx.x * blockDim.x + threadIdx.x;
    if (i < n) p[i] = val;
}

// ---------------------------------------------------------------------------
// out[n,CC] = A[n,K] * W[K,CC] + bias  via v_wmma_f32_16x16x4_f32.
// One wave = one 16x16 output tile; 8 waves / 256-thread block.
// n must be a multiple of 16 (N = 100000 = 16*6250).  K in {128, 64}.
// CP = CC padded up to a multiple of 16; W is staged TRANSPOSED and
// zero-padded into LDS (sWt[c*K + k]) so B-fragment loads are a single
// ds_load_b64 and the unrolled inner loop has no divergence.
//
// ISA fragment layouts (wave32):
//   A (16x4 f32): lanes 0-15 hold M=lane, K=k0+{0,1}; lanes 16-31 K=k0+{2,3}.
//   B (4x16):     lanes 0-15 hold N=lane, K=k0+{0,1}; lanes 16-31 K=k0+{2,3}.
//   C/D:          VGPR r holds M = r + (lane<16 ? 0 : 8), N = lane&15.
// ---------------------------------------------------------------------------
template <int K, int CC, int CP>
__global__ __launch_bounds__(256) void gat_gemm_bias_f32_wmma(
    const float* __restrict__ A, const float* __restrict__ W,
    const float* __restrict__ bias, float* __restrict__ out, int n) {
    __shared__ float sWt[CP * K];  // W^T, zero padded: sWt[c*K + k]

    // cooperative staging of W^T into LDS
    for (int idx = threadIdx.x; idx < CP * K; idx += blockDim.x) {
        const int c = idx / K;
        const int k = idx % K;
        sWt[idx] = (c < CC) ? W[(size_t)k * CC + c] : 0.0f;
    }
    __syncthreads();

    const int waveId = threadIdx.x >> 5;
    const int lane = threadIdx.x & 31;
    constexpr int colTiles = CP >> 4;
    const int rowTiles = n >> 4;
    const int tileId = blockIdx.x * (blockDim.x >> 5) + waveId;
    const int rowTile = tileId / colTiles;
    const int colTile = tileId % colTiles;
    if (rowTile >= rowTiles) return;  // wave-uniform; EXEC all-1 for WMMA

    const int mrow = (rowTile << 4) + (lane & 15);
    const int colp = (colTile << 4) + (lane & 15);
    const int khalf = (lane >> 4) << 1;  // 0 or 2

    const float* __restrict__ aRow = A + (size_t)mrow * K + khalf;
    const float* __restrict__ bCol = sWt + (size_t)colp * K + khalf;

    v8f acc = {};
#pragma unroll
    for (int k0 = 0; k0 < K; k0 += 4) {
        const v2f a = *(const v2f*)(aRow + k0);  // 8B-aligned global_load_b64
        const v2f b = *(const v2f*)(bCol + k0);  // ds_load_b64
        acc = __builtin_amdgcn_wmma_f32_16x16x4_f32(
            false, a, false, b, (short)0, acc, false, false);
    }

    if (colp < CC) {
        const float bv = bias[colp];
        const int rbase = (rowTile << 4) + ((lane >> 4) << 3);
#pragma unroll
        for (int r = 0; r < 8; ++r) {
            out[(size_t)(rbase + r) * CC + colp] = acc[r] + bv;
        }
    }
}

// ---------------------------------------------------------------------------
// Layer 1 edge passes (HEADS=8, DIM_H=8): one thread per (edge, head)
// ---------------------------------------------------------------------------
__global__ void gat_edge_logits_max_l1(
    const float* __restrict__ xl, const float* __restrict__ xr,
    const float* __restrict__ att, const int* __restrict__ src,
    const int* __restrict__ dst, float* __restrict__ e1,
    float* __restrict__ m1, long long E8) {
    long long t = (long long)blockIdx.x * blockDim.x + threadIdx.x;
    if (t >= E8) return;
    const int e = (int)(t >> 3);
    const int h = (int)(t & 7);
    const int s = src[e], d = dst[e];
    const float* pj = xl + (size_t)s * HID + h * DIM_H;
    const float* pi = xr + (size_t)d * HID + h * DIM_H;
    const float* pa = att + h * DIM_H;
    float acc = 0.0f;
#pragma unroll
    for (int c = 0; c < DIM_H; ++c) {
        float v = pi[c] + pj[c];
        v = v > 0.0f ? v : v * NEG_SLOPE;
        acc += v * pa[c];
    }
    e1[t] = acc;
    atomicMaxFloat(&m1[(size_t)d * HEADS + h], acc);
}

__global__ void gat_edge_expsum_l1(
    const int* __restrict__ dst, const float* __restrict__ m1,
    float* __restrict__ e1, float* __restrict__ denom1, long long E8) {
    long long t = (long long)blockIdx.x * blockDim.x + threadIdx.x;
    if (t >= E8) return;
    const int e = (int)(t >> 3);
    const int h = (int)(t & 7);
    const int d = dst[e];
    const float ex = expf(e1[t] - m1[(size_t)d * HEADS + h]);
    e1[t] = ex;
    atomicAdd(&denom1[(size_t)d * HEADS + h], ex);
}

__global__ void gat_edge_aggregate_l1(
    const int* __restrict__ src, const int* __restrict__ dst,
    const float* __restrict__ xl, const float* __restrict__ e1,
    const float* __restrict__ denom1, float* __restrict__ out1, long long E8) {
    long long t = (long long)blockIdx.x * blockDim.x + threadIdx.x;
    if (t >= E8) return;
    const int e = (int)(t >> 3);
    const int h = (int)(t & 7);
    const int s = src[e], d = dst[e];
    const float alpha = e1[t] / denom1[(size_t)d * HEADS + h];
    const float* pj = xl + (size_t)s * HID + h * DIM_H;
    float* po = out1 + (size_t)d * HID + h * DIM_H;
#pragma unroll
    for (int c = 0; c < DIM_H; ++c) {
        atomicAdd(&po[c], pj[c] * alpha);
    }
}

// out1 = ELU(out1 + bias1), in place (becomes layer-2 input h1)
__global__ void gat_bias_elu(float* __restrict__ out1,
                             const float* __restrict__ bias1, long long total) {
    long long i = (long long)blockIdx.x * blockDim.x + threadIdx.x;
    if (i >= total) return;
    float v = out1[i] + bias1[(int)(i & (HID - 1))];
    out1[i] = v > 0.0f ? v : expm1f(v);
}

// ---------------------------------------------------------------------------
// Layer 2 edge passes (1 head, 40 channels): 8 lanes per edge + shfl reduce
// ---------------------------------------------------------------------------
__global__ void gat_edge_logits_max_l2(
    const float* __restrict__ xl, const float* __restrict__ xr,
    const float* __restrict__ att, const int* __restrict__ src,
    const int* __restrict__ dst, float* __restrict__ e2,
    float* __restrict__ m2, long long E8) {
    long long t = (long long)blockIdx.x * blockDim.x + threadIdx.x;
    if (t >= E8) return;
    const int e = (int)(t >> 3);
    const int sub = (int)(t & 7);
    const int s = src[e], d = dst[e];
    const float* pj = xl + (size_t)s * DIM_OUT;
    const float* pi = xr + (size_t)d * DIM_OUT;
    float acc = 0.0f;
#pragma unroll
    for (int c = sub; c < DIM_OUT; c += 8) {
        float v = pi[c] + pj[c];
        v = v > 0.0f ? v : v * NEG_SLOPE;
        acc += v * att[c];
    }
    // reduce within the contiguous 8-lane group of the wave32
    acc += __shfl_xor(acc, 1, 32);
    acc += __shfl_xor(acc, 2, 32);
    acc += __shfl_xor(acc, 4, 32);
    if (sub == 0) {
        e2[e] = acc;
        atomicMaxFloat(&m2[d], acc);
    }
}

__global__ void gat_edge_expsum_l2(
    const int* __restrict__ dst, const float* __restrict__ m2,
    float* __restrict__ e2, float* __restrict__ denom2, int E) {
    int e = blockIdx.x * blockDim.x + threadIdx.x;
    if (e >= E) return;
    const int d = dst[e];
    const float ex = expf(e2[e] - m2[d]);
    e2[e] = ex;
    atomicAdd(&denom2[d], ex);
}

__global__ void gat_edge_aggregate_l2(
    const int* __restrict__ src, const int* __restrict__ dst,
    const float* __restrict__ xl, const float* __restrict__ e2,
    const float* __restrict__ denom2, float* __restrict__ out2, long long E8) {
    long long t = (long long)blockIdx.x * blockDim.x + threadIdx.x;
    if (t >= E8) return;
    const int e = (int)(t >> 3);
    const int sub = (int)(t & 7);
    const int s = src[e], d = dst[e];
    const float alpha = e2[e] / denom2[d];
    const float* pj = xl + (size_t)s * DIM_OUT;
    float* po = out2 + (size_t)d * DIM_OUT;
#pragma unroll
    for (int c = sub; c < DIM_OUT; c += 8) {
        atomicAdd(&po[c], pj[c] * alpha);
    }
}

// ---------------------------------------------------------------------------
// finalize: h2 = out2 + bias2 ; d_out[0:N*40] = h2 ; d_out[N*40:] = logsoftmax
// one wave (32 lanes) per node; lanes 0-31 hold c=lane, lanes 0-7 also c=32+lane
// ---------------------------------------------------------------------------
__global__ __launch_bounds__(256) void gat_finalize(
    const float* __restrict__ out2, const float* __restrict__ bias2,
    float* __restrict__ d_out, int N) {
    const int node = blockIdx.x * (blockDim.x >> 5) + (threadIdx.x >> 5);
    const int lane = threadIdx.x & 31;
    if (node >= N) return;
    const float* row = out2 + (size_t)node * DIM_OUT;
    const float a = row[lane] + bias2[lane];
    const bool has2 = (lane < DIM_OUT - 32);
    const float b = has2 ? (row[32 + lane] + bias2[32 + lane]) : -__builtin_inff();
    float mx = fmaxf(a, b);
#pragma unroll
    for (int o = 16; o >= 1; o >>= 1) mx = fmaxf(mx, __shfl_xor(mx, o, 32));
    float s = expf(a - mx) + (has2 ? expf(b - mx) : 0.0f);
#pragma unroll
    for (int o = 16; o >= 1; o >>= 1) s += __shfl_xor(s, o, 32);
    const float lse = logf(s) + mx;
    float* h  = d_out + (size_t)node * DIM_OUT;
    float* ls = d_out + (size_t)N * DIM_OUT + (size_t)node * DIM_OUT;
    h[lane] = a;
    ls[lane] = a - lse;
    if (has2) {
        h[32 + lane] = b;
        ls[32 + lane] = b - lse;
    }
}

// ---------------------------------------------------------------------------
static inline int cdiv_i(long long a, long long b) { return (int)((a + b - 1) / b); }

extern "C" void kernel_launch(void* const* d_in, const int* in_sizes, int n_in,
                              void* d_out, int out_size, void* d_ws, size_t ws_size,
                              hipStream_t stream) {
    (void)n_in; (void)out_size; (void)ws_size;
    // inputs in setup_inputs() order
    const float* x     = (const float*)d_in[0];
    const int*   eidx  = (const int*)d_in[1];
    const float* Wl1   = (const float*)d_in[2];
    const float* bl1   = (const float*)d_in[3];
    const float* Wr1   = (const float*)d_in[4];
    const float* br1   = (const float*)d_in[5];
    const float* att1  = (const float*)d_in[6];
    const float* bias1 = (const float*)d_in[7];
    const float* Wl2   = (const float*)d_in[8];
    const float* bl2   = (const float*)d_in[9];
    const float* Wr2   = (const float*)d_in[10];
    const float* br2   = (const float*)d_in[11];
    const float* att2  = (const float*)d_in[12];
    const float* bias2 = (const float*)d_in[13];

    const int N = in_sizes[0] / DIM_IN;   // 100000 (multiple of 16)
    const int E = in_sizes[1] / 2;        // 1600000
    const int* src = eidx;
    const int* dst = eidx + E;

    // workspace layout (floats)
    float* ws = (float*)d_ws;
    size_t o = 0;
    float* h_l1   = ws + o; o += (size_t)N * HID;
    float* h_r1   = ws + o; o += (size_t)N * HID;
    float* e1     = ws + o; o += (size_t)E * HEADS;
    float* m1     = ws + o; o += (size_t)N * HEADS;
    float* denom1 = ws + o; o += (size_t)N * HEADS;
    float* out1   = ws + o; o += (size_t)N * HID;   // becomes h1 after ELU
    // layer-2 buffers reuse the h_l1/h_r1/e1 region (free after layer-1 agg)
    float* h_l2   = ws;
    float* h_r2   = h_l2 + (size_t)N * DIM_OUT;
    float* e2     = h_r2 + (size_t)N * DIM_OUT;
    float* out2   = e2 + (size_t)E;
    float* m2     = out2 + (size_t)N * DIM_OUT;
    float* denom2 = m2 + (size_t)N;

    const int B = 256;
    const long long E8 = (long long)E * 8;
    const unsigned int NEG_INF_BITS = 0xFF800000u;

    // ---------------- layer 1 ----------------
    gat_fill_u32<<<cdiv_i((long long)N * HEADS, B), B, 0, stream>>>(
        (unsigned int*)m1, NEG_INF_BITS, (long long)N * HEADS);
    gat_fill_u32<<<cdiv_i((long long)N * HEADS, B), B, 0, stream>>>(
        (unsigned int*)denom1, 0u, (long long)N * HEADS);
    gat_fill_u32<<<cdiv_i((long long)N * HID, B), B, 0, stream>>>(
        (unsigned int*)out1, 0u, (long long)N * HID);

    {
        const int tiles = (N >> 4) * (HID >> 4);        // CP = 64
        const int grid = cdiv_i(tiles, B / 32);
        gat_gemm_bias_f32_wmma<DIM_IN, HID, HID>
            <<<grid, B, 0, stream>>>(x, Wl1, bl1, h_l1, N);
        gat_gemm_bias_f32_wmma<DIM_IN, HID, HID>
            <<<grid, B, 0, stream>>>(x, Wr1, br1, h_r1, N);
    }

    gat_edge_logits_max_l1<<<cdiv_i(E8, B), B, 0, stream>>>(
        h_l1, h_r1, att1, src, dst, e1, m1, E8);
    gat_edge_expsum_l1<<<cdiv_i(E8, B), B, 0, stream>>>(dst, m1, e1, denom1, E8);
    gat_edge_aggregate_l1<<<cdiv_i(E8, B), B, 0, stream>>>(
        src, dst, h_l1, e1, denom1, out1, E8);
    gat_bias_elu<<<cdiv_i((long long)N * HID, B), B, 0, stream>>>(
        out1, bias1, (long long)N * HID);

    // ---------------- layer 2 ----------------
    gat_fill_u32<<<cdiv_i(N, B), B, 0, stream>>>((unsigned int*)m2, NEG_INF_BITS, N);
    gat_fill_u32<<<cdiv_i(N, B), B, 0, stream>>>((unsigned int*)denom2, 0u, N);
    gat_fill_u32<<<cdiv_i((long long)N * DIM_OUT, B), B, 0, stream>>>(
        (unsigned int*)out2, 0u, (long long)N * DIM_OUT);

    {
        const int tiles = (N >> 4) * (48 >> 4);         // CP = 48
        const int grid = cdiv_i(tiles, B / 32);
        gat_gemm_bias_f32_wmma<HID, DIM_OUT, 48>
            <<<grid, B, 0, stream>>>(out1, Wl2, bl2, h_l2, N);
        gat_gemm_bias_f32_wmma<HID, DIM_OUT, 48>
            <<<grid, B, 0, stream>>>(out1, Wr2, br2, h_r2, N);
    }

    gat_edge_logits_max_l2<<<cdiv_i(E8, B), B, 0, stream>>>(
        h_l2, h_r2, att2, src, dst, e2, m2, E8);
    gat_edge_expsum_l2<<<cdiv_i(E, B), B, 0, stream>>>(dst, m2, e2, denom2, E);
    gat_edge_aggregate_l2<<<cdiv_i(E8, B), B, 0, stream>>>(
        src, dst, h_l2, e2, denom2, out2, E8);

    gat_finalize<<<cdiv_i(N, B / 32), B, 0, stream>>>((const float*)out2, bias2,
                                                      (float*)d_out, N);
}